// ForcingFunction_15504831938706
// MI455X (gfx1250) — compile-verified
//
#include <hip/hip_runtime.h>

// Problem shape (from reference setup_inputs): B=32, H=W=512, C=1, fp32.
static constexpr int   Hh  = 512;
static constexpr int   Ww  = 512;
static constexpr float EPSC = 1e-7f;
static constexpr float DTC  = 0.05f;

// ---------------------------------------------------------------------------
// CDNA5 async global->LDS copy (ASYNCcnt path, CDNA5 ISA).
// Per-lane: copies 16 bytes (B128) from the global address in vaddr to the LDS address in vdst. EXEC-masked, so
// issuing inside a divergent `if` predicates lanes correctly.
// ---------------------------------------------------------------------------
__device__ __forceinline__ unsigned lds_lo(const void* p) {
    // Generic pointer to LDS = {shared aperture hi32, 32-bit LDS offset}; the
    // low 32 bits are exactly the wave-relative LDS byte address the async
    // instruction's VDST operand expects.
    return (unsigned)(unsigned long long)p;
}

__device__ __forceinline__ void async_load_b128(unsigned lds_addr, const float* gptr) {
    asm volatile("global_load_async_to_lds_b128 %0, %1, off"
                 :: "v"(lds_addr), "v"(gptr)
                 : "memory");
}

__device__ __forceinline__ void wait_async_all() {
    asm volatile("s_wait_asynccnt 0" ::: "memory");
}

// ---------------------------------------------------------------------------
// Kernel 0: zero the global-max accumulator (d_ws is poisoned with 0xAA).
// ---------------------------------------------------------------------------
__global__ void k_init_max(unsigned* __restrict__ gmax) {
    if (threadIdx.x == 0 && blockIdx.x == 0) *gmax = 0u;
}

// ---------------------------------------------------------------------------
// Kernel 1: raw = (ex^2 + ey^2)^2 (Sobel, zero-padded SAME) and global max.
// raw >= 0, so IEEE float ordering == unsigned-int ordering of the bits:
// atomicMax on the uint view is a deterministic float max.
// ---------------------------------------------------------------------------
__global__ __launch_bounds__(256) void k_edge_max(const float* __restrict__ img,
                                                  unsigned* __restrict__ gmax) {
    size_t idx  = (size_t)blockIdx.x * 256 + threadIdx.x;
    int    x    = (int)(idx % Ww);
    int    y    = (int)((idx / Ww) % Hh);
    size_t base = (idx / ((size_t)Ww * Hh)) * (size_t)Ww * Hh;

    auto ld = [&](int yy, int xx) -> float {
        if ((unsigned)yy < (unsigned)Hh && (unsigned)xx < (unsigned)Ww)
            return img[base + (size_t)yy * Ww + xx];
        return 0.0f;
    };

    float a00 = ld(y - 1, x - 1), a01 = ld(y - 1, x), a02 = ld(y - 1, x + 1);
    float a10 = ld(y,     x - 1),                     a12 = ld(y,     x + 1);
    float a20 = ld(y + 1, x - 1), a21 = ld(y + 1, x), a22 = ld(y + 1, x + 1);

    // KLX cross-correlation
    float ex = (a00 - a02) + 2.0f * (a10 - a12) + (a20 - a22);
    // KLY cross-correlation
    float ey = (a00 + 2.0f * a01 + a02) - (a20 + 2.0f * a21 + a22);
    float g2  = ex * ex + ey * ey;
    float raw = g2 * g2;

    __shared__ float red[256];
    red[threadIdx.x] = raw;
    __syncthreads();
    for (int s = 128; s > 0; s >>= 1) {
        if ((int)threadIdx.x < s)
            red[threadIdx.x] = fmaxf(red[threadIdx.x], red[threadIdx.x + s]);
        __syncthreads();
    }
    if (threadIdx.x == 0) atomicMax(gmax, __float_as_uint(red[0]));
}

// ---------------------------------------------------------------------------
// Kernel 2: fused forcing function. 32x32 output tile / 256 threads (8 waves).
// u/img staged to LDS via async B128 copies (halo 2 in y, halo 4 in x so every
// chunk is 16B aligned and every 4-float chunk is wholly in- or out-of-image).
// edges e = 1/(raw/M + 1), px = ux/(|grad u|+eps), py likewise, and |grad u|
// are computed ONCE per (tile+halo1) point in LDS, then combined per pixel.
// ---------------------------------------------------------------------------
__global__ __launch_bounds__(256) void k_forcing(
    const float* __restrict__ u,       const float* __restrict__ img,
    const float* __restrict__ alpha_p, const float* __restrict__ beta_p,
    const float* __restrict__ gamma_p, const unsigned* __restrict__ gmax,
    float* __restrict__ out)
{
    constexpr int TS = 32;            // output tile edge
    constexpr int UX = 40, UY = 36;   // u/img LDS tile (x halo 4, y halo 2)
    constexpr int ES = 34, EP = 36;   // derived-field tile 34x34, pitch 36

    // alignas(16): async B128 writes 16B/lane into LDS; chunk offsets are
    // multiples of 16B, so the array base must be 16B aligned too.
    __shared__ alignas(16) float u_s [UY * UX];
    __shared__ alignas(16) float i_s [UY * UX];
    __shared__ float e_s [ES * EP];
    __shared__ float px_s[ES * EP];
    __shared__ float py_s[ES * EP];
    __shared__ float nm_s[ES * EP];

    const int    tid  = threadIdx.x;
    const int    x0   = blockIdx.x * TS;
    const int    y0   = blockIdx.y * TS;
    const size_t base = (size_t)blockIdx.z * (size_t)Hh * Ww;
    const float* ub   = u + base;
    const float* ib   = img + base;

    // Boundary blocks: pre-zero LDS so out-of-image samples read as 0
    // (implements the zero padding of conv(..., 'SAME') on u and img).
    const bool interior = (x0 >= 4) && (x0 + 36 <= Ww) && (y0 >= 2) && (y0 + 34 <= Hh);
    if (!interior) {
        for (int i = tid; i < UY * UX; i += 256) { u_s[i] = 0.0f; i_s[i] = 0.0f; }
        __syncthreads();   // zeros visible before async writes land
    }

    // Async DMA: 36 rows x 10 aligned 16B chunks per array. Chunks are
    // entirely in- or out-of-image (W % 4 == 0, chunk start = x0-4+4c).
    for (int i = tid; i < UY * 10; i += 256) {
        int r  = i / 10;
        int c  = i - r * 10;
        int gy = y0 - 2 + r;
        int gx = x0 - 4 + c * 4;
        if ((unsigned)gy < (unsigned)Hh && (unsigned)gx < (unsigned)Ww) {
            size_t off = (size_t)gy * Ww + gx;
            int    l   = r * UX + c * 4;
            async_load_b128(lds_lo(&u_s[l]), ub + off);
            async_load_b128(lds_lo(&i_s[l]), ib + off);
        }
    }
    wait_async_all();      // this wave's async copies committed to LDS
    __syncthreads();       // all waves' copies visible

    const float M    = __uint_as_float(*gmax);
    const float invM = 1.0f / M;

    // Stage 2: derived fields on tile + halo 1 (34x34). Fields sampled
    // outside the image must be 0 (conv zero-pads the *field*, not u/img).
    for (int i = tid; i < ES * ES; i += 256) {
        int rr = i / ES;
        int cc = i - rr * ES;
        int gx = x0 - 1 + cc;
        int gy = y0 - 1 + rr;
        float e = 0.0f, px = 0.0f, py = 0.0f, nm = 0.0f;
        if ((unsigned)gx < (unsigned)Ww && (unsigned)gy < (unsigned)Hh) {
            int lx = cc + 3, ly = rr + 1;
            const float* c0 = &i_s[(ly - 1) * UX + lx];
            const float* c1 = &i_s[ ly      * UX + lx];
            const float* c2 = &i_s[(ly + 1) * UX + lx];
            float ex  = (c0[-1] - c0[1]) + 2.0f * (c1[-1] - c1[1]) + (c2[-1] - c2[1]);
            float ey  = (c0[-1] + 2.0f * c0[0] + c0[1]) - (c2[-1] + 2.0f * c2[0] + c2[1]);
            float g2  = ex * ex + ey * ey;
            float raw = g2 * g2;
            // edges = 1 / (raw/M + 1)
            e = 1.0f / (raw * invM + 1.0f);

            float gux = u_s[ly * UX + lx + 1] - u_s[ly * UX + lx - 1];     // KXC
            float guy = u_s[(ly - 1) * UX + lx] - u_s[(ly + 1) * UX + lx]; // KYC
            nm = sqrtf(gux * gux + guy * guy + EPSC);
            float d = nm + EPSC;
            px = gux / d;
            py = guy / d;
        }
        e_s [rr * EP + cc] = e;
        px_s[rr * EP + cc] = px;
        py_s[rr * EP + cc] = py;
        nm_s[rr * EP + cc] = nm;
    }
    __syncthreads();

    const float alpha = alpha_p[0];
    const float beta  = beta_p[0];
    const float gamma = gamma_p[0];

    // Stage 3: per-pixel combination, coalesced output stores.
    for (int i = tid; i < TS * TS; i += 256) {
        int r  = i >> 5;
        int c  = i & 31;
        int lx = c + 4, ly = r + 2;   // u/img tile coords
        int ec = c + 1, er = r + 1;   // derived-field tile coords

        float ucc = u_s[ly * UX + lx];
        float xp  = u_s[ly * UX + lx + 1] - ucc;        // KXP
        float xn  = ucc - u_s[ly * UX + lx - 1];        // KXN
        float yp  = u_s[(ly - 1) * UX + lx] - ucc;      // KYP
        float yn  = ucc - u_s[(ly + 1) * UX + lx];      // KYN

        float gex = e_s[er * EP + ec + 1] - e_s[er * EP + ec - 1];     // KXC on edges
        float gey = e_s[(er - 1) * EP + ec] - e_s[(er + 1) * EP + ec]; // KYC on edges
        float fxp = fmaxf(gex, 0.0f), fxn = fminf(gex, 0.0f);
        float fyp = fmaxf(gey, 0.0f), fyn = fminf(gey, 0.0f);
        float transport = 20.0f * ((fxp * xp + fxn * xn) + (fyp * yp + fyn * yn));

        float kappa = (px_s[er * EP + ec + 1] - px_s[er * EP + ec - 1])
                    + (py_s[(er - 1) * EP + ec] - py_s[(er + 1) * EP + ec]);

        float ee = e_s [er * EP + ec];
        float nm = nm_s[er * EP + ec];
        float curvature = ee * kappa * nm;
        float balloon   = ee * nm;

        out[base + (size_t)(y0 + r) * Ww + (x0 + c)] =
            ucc + DTC * (curvature * alpha + transport * beta + balloon * gamma);
    }
}

// ---------------------------------------------------------------------------
// Host entry
// ---------------------------------------------------------------------------
extern "C" void kernel_launch(void* const* d_in, const int* in_sizes, int n_in,
                              void* d_out, int out_size, void* d_ws, size_t ws_size,
                              hipStream_t stream) {
    (void)in_sizes; (void)n_in; (void)out_size; (void)ws_size;
    const float* u     = (const float*)d_in[0];
    const float* img   = (const float*)d_in[1];
    const float* alpha = (const float*)d_in[2];
    const float* beta  = (const float*)d_in[3];
    const float* gamma = (const float*)d_in[4];
    float*       out   = (float*)d_out;
    unsigned*    gmax  = (unsigned*)d_ws;

    k_init_max<<<1, 1, 0, stream>>>(gmax);

    const int nblocksA = (32 * Hh * Ww) / 256;   // 32768
    k_edge_max<<<nblocksA, 256, 0, stream>>>(img, gmax);

    dim3 gridB(Ww / 32, Hh / 32, 32);            // 16 x 16 x 32
    k_forcing<<<gridB, 256, 0, stream>>>(u, img, alpha, beta, gamma, gmax, out);
}